// RWKV_Tmix_x070_Mose_cxa078_38560216383805
// MI455X (gfx1250) — compile-verified
//
#include <hip/hip_runtime.h>
#include <cmath>

typedef __attribute__((ext_vector_type(2))) float v2f;
typedef __attribute__((ext_vector_type(8))) float v8f;

constexpr int Bq = 4, Tq = 1024, Cq = 1024;
constexpr int Hq = 16, HKVq = 8, Nq = 64;
constexpr int MT = Bq * Tq;                 // 4096 token rows
constexpr float EPSq = 1e-6f;
constexpr int RG = 4;                       // row-groups per wkv7 chain

constexpr int EP_NONE = 0, EP_TANH = 1, EP_SIG = 2, EP_SIGB = 3, EP_DECAYB = 4;

__device__ __forceinline__ float sigf(float x) { return 1.0f / (1.0f + expf(-x)); }

__device__ __forceinline__ float wave_sum32(float v) {
#pragma unroll
  for (int d = 16; d > 0; d >>= 1) v += __shfl_xor(v, d, 32);
  return v;
}

// ---------------------------------------------------------------------------
// Generic fp32 WMMA GEMM:  C[M,Ncols] = epilogue(A[M,K] @ W[K,Ncols] (+bias))
// One wave (32 threads) computes a 16 x (16*NT) output tile via
// v_wmma_f32_16x16x4_f32 (exact fp32), K stepped by 4.
// Fragment layouts per CDNA5 ISA 7.12.2:
//   A 16x4 : lane(0..15)=M, K = vgprIdx + 2*(lane>>4)
//   B 4x16 : lane&15 = N,  K = vgprIdx + 2*(lane>>4)
//   C 16x16: lane&15 = N,  M = vgprIdx + 8*(lane>>4)
// ---------------------------------------------------------------------------
template <int NT, int MODE>
__global__ __launch_bounds__(32) void gemm_wmma_f32(
    const float* __restrict__ A, const float* __restrict__ W,
    const float* __restrict__ bias, float* __restrict__ C,
    int K, int Ncols) {
  const int lane = threadIdx.x & 31;
  const int mr = lane & 15;
  const int kg = lane >> 4;
  const int m0 = blockIdx.y << 4;
  const int n0 = blockIdx.x * (16 * NT);

  v8f acc[NT];
#pragma unroll
  for (int t = 0; t < NT; ++t)
#pragma unroll
    for (int v = 0; v < 8; ++v) acc[t][v] = 0.0f;

  const float* Arow = A + (size_t)(m0 + mr) * K;

  for (int k0 = 0; k0 < K; k0 += 4) {
    const int ka = k0 + (kg << 1);
    v2f a;
    a.x = Arow[ka];
    a.y = Arow[ka + 1];
    const float* Wp = W + (size_t)ka * Ncols + n0 + mr;
#pragma unroll
    for (int t = 0; t < NT; ++t) {
      v2f b;
      b.x = Wp[t * 16];
      b.y = Wp[t * 16 + Ncols];
      acc[t] = __builtin_amdgcn_wmma_f32_16x16x4_f32(
          false, a, false, b, (short)0, acc[t], false, false);
    }
  }

#pragma unroll
  for (int t = 0; t < NT; ++t) {
    const int n = n0 + t * 16 + mr;
    float bv = 0.0f;
    if (MODE == EP_SIGB || MODE == EP_DECAYB) bv = bias[n];
#pragma unroll
    for (int v = 0; v < 8; ++v) {
      const int m = m0 + v + (kg << 3);
      float x = acc[t][v];
      if (MODE == EP_TANH) x = tanhf(x);
      else if (MODE == EP_SIG) x = sigf(x);
      else if (MODE == EP_SIGB) x = sigf(x + bv);
      else if (MODE == EP_DECAYB)
        // w = -softplus(-(w0+z)) - 0.5 ; dec = exp(-exp(w)) = exp(-e^-0.5 * sigmoid(w0+z))
        x = expf(-0.6065306597126334f * sigf(x + bv));
      C[(size_t)m * Ncols + n] = x;
    }
  }
}

// ---------------------------------------------------------------------------
// Per-(b,t,h) prep: head RMS norm, RoPE, GQA repeat, v_first mixing,
// k-hat normalization; emits r (in place), k, v, -kk, kk*a.
// One wave per (b,t,h); lane handles n = lane and n+32.
// ---------------------------------------------------------------------------
__global__ __launch_bounds__(256) void prep_kernel(
    float* __restrict__ R,            // in: r_lin [B*T, H*N]; out in place
    const float* __restrict__ Klin,   // [B*T, HKV*N]
    const float* __restrict__ Vlin,   // [B*T, HKV*N]
    const float* __restrict__ vfirst, // [B*T, C]
    const float* __restrict__ MIXp,   // [B*T, C] sigmoid already applied
    const float* __restrict__ ASIGp,  // [B*T, C] sigmoid already applied
    const float* __restrict__ rnw, const float* __restrict__ knw,
    float* __restrict__ Kout, float* __restrict__ Vout,
    float* __restrict__ Aout, float* __restrict__ Bout) {
  const int lane = threadIdx.x & 31;
  const int gid = blockIdx.x * 8 + (threadIdx.x >> 5); // (b*T+t)*H + h
  const int h = gid % Hq;
  const int bt = gid / Hq;
  const int t = bt % Tq;
  const int hkv = h >> 1; // REP = 2

  const size_t off = (size_t)gid * Nq;
  const size_t offk = ((size_t)bt * HKVq + hkv) * Nq;

  float r0 = R[off + lane], r1 = R[off + lane + 32];
  float k0 = Klin[offk + lane], k1 = Klin[offk + lane + 32];

  float s = rsqrtf(wave_sum32(r0 * r0 + r1 * r1) * (1.0f / 64.0f) + EPSq);
  r0 *= s * rnw[lane];
  r1 *= s * rnw[lane + 32];

  s = rsqrtf(wave_sum32(k0 * k0 + k1 * k1) * (1.0f / 64.0f) + EPSq);
  k0 *= s * knw[lane];
  k1 *= s * knw[lane + 32];

  // RoPE: half = 32, angle(n) = t * theta^(-(n%32)/32), same for n and n+32
  const float ang = (float)t * powf(1.0e6f, -(float)lane * (1.0f / 32.0f));
  float sn, cs;
  sincosf(ang, &sn, &cs);
  const float rn0 = r0 * cs - r1 * sn, rn1 = r1 * cs + r0 * sn;
  const float kn0 = k0 * cs - k1 * sn, kn1 = k1 * cs + k0 * sn;

  const float nk = sqrtf(wave_sum32(kn0 * kn0 + kn1 * kn1));
  const float inv = 1.0f / fmaxf(nk, 1e-12f);
  const float kk0 = kn0 * inv, kk1 = kn1 * inv;

  const float vv0 = Vlin[offk + lane], vv1 = Vlin[offk + lane + 32];
  const float mx0 = MIXp[off + lane], mx1 = MIXp[off + lane + 32];
  const float vf0 = vfirst[off + lane], vf1 = vfirst[off + lane + 32];
  const float v0 = vv0 + (vf0 - vv0) * mx0;
  const float v1 = vv1 + (vf1 - vv1) * mx1;
  const float a0 = ASIGp[off + lane], a1 = ASIGp[off + lane + 32];

  R[off + lane] = rn0;
  R[off + lane + 32] = rn1;
  Kout[off + lane] = kn0;
  Kout[off + lane + 32] = kn1;
  Vout[off + lane] = v0;
  Vout[off + lane + 32] = v1;
  Aout[off + lane] = -kk0;
  Aout[off + lane + 32] = -kk1;
  Bout[off + lane] = kk0 * a0;
  Bout[off + lane + 32] = kk1 * a1;
}

// ---------------------------------------------------------------------------
// wkv7 sequential scan, row-split. State rows are independent:
//   sa_i = sum_j S[i][j]*a[j];  S[i][j] = S[i][j]*w[j] + sa_i*b[j] + v[i]*k[j]
//   y[i] = sum_j S[i][j]*r[j]
// One block per (chain, row-group): grid = B*H*RG = 256 blocks of 64 threads
// (2 waves). Each block owns 16 rows; 4 lanes per row, 16 cols per lane in
// registers. Per-step vectors staged through LDS with register prefetch of
// step t+1 issued before the compute of step t (global latency hidden).
// ---------------------------------------------------------------------------
__global__ __launch_bounds__(64) void wkv7_scan_kernel(
    const float* __restrict__ R, const float* __restrict__ Wd,
    const float* __restrict__ K, const float* __restrict__ V,
    const float* __restrict__ A, const float* __restrict__ Bc,
    float* __restrict__ Y) {
  const int chain = blockIdx.x >> 2;   // b*H + h
  const int rg = blockIdx.x & (RG - 1);
  const int tid = threadIdx.x;         // 0..63
  const int rloc = tid >> 2;           // local row 0..15
  const int cb = (tid & 3) << 4;       // key-column base

  __shared__ float lr[64], lw[64], lk[64], la[64], lb[64], lv[16], ly[16];

  float sS[16];
#pragma unroll
  for (int j = 0; j < 16; ++j) sS[j] = 0.0f;

  const int b = chain / Hq, h = chain % Hq;
  size_t off = ((size_t)(b * Tq) * Hq + h) * Nq;
  const size_t stride = (size_t)Hq * Nq;

  // prefetch step 0 into registers
  float pr = R[off + tid], pw = Wd[off + tid], pk = K[off + tid],
        pa = A[off + tid], pb = Bc[off + tid];
  float pv = (tid < 16) ? V[off + (rg << 4) + tid] : 0.0f;

  for (int t = 0; t < Tq; ++t) {
    lr[tid] = pr; lw[tid] = pw; lk[tid] = pk; la[tid] = pa; lb[tid] = pb;
    if (tid < 16) lv[tid] = pv;
    __syncthreads();

    if (t + 1 < Tq) { // prefetch next step while computing this one
      const size_t o2 = off + stride;
      pr = R[o2 + tid]; pw = Wd[o2 + tid]; pk = K[o2 + tid];
      pa = A[o2 + tid]; pb = Bc[o2 + tid];
      if (tid < 16) pv = V[o2 + (rg << 4) + tid];
    }

    float sa = 0.0f;
#pragma unroll
    for (int j = 0; j < 16; ++j) sa += sS[j] * la[cb + j];
    sa += __shfl_xor(sa, 1, 32);
    sa += __shfl_xor(sa, 2, 32);

    const float vi = lv[rloc];
    float y = 0.0f;
#pragma unroll
    for (int j = 0; j < 16; ++j) {
      const float sj = sS[j] * lw[cb + j] + sa * lb[cb + j] + vi * lk[cb + j];
      sS[j] = sj;
      y += sj * lr[cb + j];
    }
    y += __shfl_xor(y, 1, 32);
    y += __shfl_xor(y, 2, 32);
    if ((tid & 3) == 0) ly[rloc] = y;
    __syncthreads();
    if (tid < 16) Y[off + (rg << 4) + tid] = ly[tid];
    off += stride;
  }
}

// ---------------------------------------------------------------------------
// y = (y + sum_n(r*k*r_k[h]) * v) * g   (in place on Y)
// ---------------------------------------------------------------------------
__global__ __launch_bounds__(256) void mixgate_kernel(
    const float* __restrict__ R, const float* __restrict__ K,
    const float* __restrict__ V, const float* __restrict__ G,
    const float* __restrict__ rk, float* __restrict__ Y) {
  const int lane = threadIdx.x & 31;
  const int gid = blockIdx.x * 8 + (threadIdx.x >> 5);
  const int h = gid % Hq;
  const size_t off = (size_t)gid * Nq;

  const float r0 = R[off + lane], r1 = R[off + lane + 32];
  const float k0 = K[off + lane], k1 = K[off + lane + 32];
  const float dot =
      wave_sum32(r0 * k0 * rk[h * Nq + lane] + r1 * k1 * rk[h * Nq + lane + 32]);

  const float y0 = Y[off + lane] + dot * V[off + lane];
  const float y1 = Y[off + lane + 32] + dot * V[off + lane + 32];
  Y[off + lane] = y0 * G[off + lane];
  Y[off + lane + 32] = y1 * G[off + lane + 32];
}

__global__ __launch_bounds__(256) void copy_f4_kernel(
    const float4* __restrict__ src, float4* __restrict__ dst, int n) {
  const int i = blockIdx.x * blockDim.x + threadIdx.x;
  if (i < n) dst[i] = src[i];
}

// ---------------------------------------------------------------------------
extern "C" void kernel_launch(void* const* d_in, const int* in_sizes, int n_in,
                              void* d_out, int out_size, void* d_ws,
                              size_t ws_size, hipStream_t stream) {
  (void)in_sizes; (void)n_in; (void)out_size; (void)ws_size;

  const float* x   = (const float*)d_in[0];
  const float* vfi = (const float*)d_in[1];
  const float* Wr  = (const float*)d_in[2];
  const float* Wk  = (const float*)d_in[3];
  const float* Wv  = (const float*)d_in[4];
  const float* Wo  = (const float*)d_in[5];
  const float* w0p = (const float*)d_in[6];
  const float* w1p = (const float*)d_in[7];
  const float* w2p = (const float*)d_in[8];
  const float* a0p = (const float*)d_in[9];
  const float* a1p = (const float*)d_in[10];
  const float* a2p = (const float*)d_in[11];
  const float* v0p = (const float*)d_in[12];
  const float* v1p = (const float*)d_in[13];
  const float* v2p = (const float*)d_in[14];
  const float* g1p = (const float*)d_in[15];
  const float* g2p = (const float*)d_in[16];
  const float* rkp = (const float*)d_in[17];
  const float* rnw = (const float*)d_in[18];
  const float* knw = (const float*)d_in[19];

  float* ws = (float*)d_ws;
  const size_t M4 = (size_t)MT * Cq; // 4,194,304 floats
  float* R    = ws + 0 * M4;
  float* Kb   = ws + 1 * M4;
  float* Vb   = ws + 2 * M4;
  float* Ab   = ws + 3 * M4;
  float* Bb   = ws + 4 * M4;
  float* DEC  = ws + 5 * M4;
  float* Yb   = ws + 6 * M4;
  float* Gg   = ws + 7 * M4;
  float* ASIG = ws + 8 * M4;
  float* MIX  = ws + 9 * M4;
  float* KLIN = ws + 10 * M4;            // MT*512
  float* VLIN = KLIN + (size_t)MT * 512;
  float* HW   = VLIN + (size_t)MT * 512; // MT*64
  float* HA   = HW + (size_t)MT * 64;    // MT*64
  float* HV   = HA + (size_t)MT * 64;    // MT*32
  float* HG   = HV + (size_t)MT * 32;    // MT*160

  const dim3 blk(32);
  const int gy = MT / 16; // 256 row tiles

  // Projections
  gemm_wmma_f32<4, EP_NONE><<<dim3(Cq / 64, gy), blk, 0, stream>>>(x, Wr, nullptr, R, Cq, Cq);
  gemm_wmma_f32<4, EP_NONE><<<dim3(512 / 64, gy), blk, 0, stream>>>(x, Wk, nullptr, KLIN, Cq, 512);
  gemm_wmma_f32<4, EP_NONE><<<dim3(512 / 64, gy), blk, 0, stream>>>(x, Wv, nullptr, VLIN, Cq, 512);
  // decay: dec = exp(-e^-0.5 * sigmoid(w0 + tanh(x@w1)@w2))
  gemm_wmma_f32<4, EP_TANH><<<dim3(1, gy), blk, 0, stream>>>(x, w1p, nullptr, HW, Cq, 64);
  gemm_wmma_f32<4, EP_DECAYB><<<dim3(Cq / 64, gy), blk, 0, stream>>>(HW, w2p, w0p, DEC, 64, Cq);
  // a = sigmoid(a0 + (x@a1)@a2)
  gemm_wmma_f32<4, EP_NONE><<<dim3(1, gy), blk, 0, stream>>>(x, a1p, nullptr, HA, Cq, 64);
  gemm_wmma_f32<4, EP_SIGB><<<dim3(Cq / 64, gy), blk, 0, stream>>>(HA, a2p, a0p, ASIG, 64, Cq);
  // mix = sigmoid(v0 + (x@v1)@v2)
  gemm_wmma_f32<2, EP_NONE><<<dim3(1, gy), blk, 0, stream>>>(x, v1p, nullptr, HV, Cq, 32);
  gemm_wmma_f32<4, EP_SIGB><<<dim3(Cq / 64, gy), blk, 0, stream>>>(HV, v2p, v0p, MIX, 32, Cq);
  // g = sigmoid(x@g1)@g2
  gemm_wmma_f32<2, EP_SIG><<<dim3(160 / 32, gy), blk, 0, stream>>>(x, g1p, nullptr, HG, Cq, 160);
  gemm_wmma_f32<4, EP_NONE><<<dim3(Cq / 64, gy), blk, 0, stream>>>(HG, g2p, nullptr, Gg, 160, Cq);

  // Per-token-head prep, row-split sequential scan, epilogue
  prep_kernel<<<(MT * Hq) / 8, 256, 0, stream>>>(R, KLIN, VLIN, vfi, MIX, ASIG,
                                                 rnw, knw, Kb, Vb, Ab, Bb);
  wkv7_scan_kernel<<<Bq * Hq * RG, 64, 0, stream>>>(R, DEC, Kb, Vb, Ab, Bb, Yb);
  mixgate_kernel<<<(MT * Hq) / 8, 256, 0, stream>>>(R, Kb, Vb, Gg, rkp, Yb);

  // out = yg @ Wo, then append v_first (tuple output)
  float* out = (float*)d_out;
  gemm_wmma_f32<4, EP_NONE><<<dim3(Cq / 64, gy), blk, 0, stream>>>(Yb, Wo, nullptr, out, Cq, Cq);
  copy_f4_kernel<<<(int)((M4 / 4) / 256), 256, 0, stream>>>(
      (const float4*)vfi, (float4*)(out + M4), (int)(M4 / 4));
}